// OutputLayer_81870666596761
// MI455X (gfx1250) — compile-verified
//
#include <hip/hip_runtime.h>
#include <math.h>

#define IN_FT   128
#define HD      64
#define NHEAD   4
#define DHEAD   16
#define SLOPE   0.2f

typedef float v2f __attribute__((ext_vector_type(2)));
typedef float v8f __attribute__((ext_vector_type(8)));

// float atomic max via monotonic int ops (works for mixed signs; init bytes 0xFF)
__device__ __forceinline__ void atomic_max_float(float* addr, float val) {
    int bits = __float_as_int(val);
    if (bits >= 0) {
        atomicMax((int*)addr, bits);
    } else {
        atomicMin((unsigned int*)addr, __float_as_uint(val));
    }
}

// ---------------------------------------------------------------------------
// feat[N,64] = h[N,128] @ W[128,64]  using V_WMMA_F32_16X16X4_F32
// grid.x = ceil(N/16) row tiles; 4 waves per block, one 16-col tile each.
// ---------------------------------------------------------------------------
__global__ __launch_bounds__(128) void feat_gemm_kernel(
    const float* __restrict__ h, const float* __restrict__ W,
    float* __restrict__ feat, int N)
{
    const int wave = threadIdx.x >> 5;      // 0..3 -> column tile
    const int lane = threadIdx.x & 31;
    const int hi   = lane >> 4;             // lane half selects K pair
    const int l    = lane & 15;
    const int row0 = blockIdx.x * 16;
    const int col0 = wave * 16;
    const bool full = (row0 + 16 <= N);

    v8f c = {0.f, 0.f, 0.f, 0.f, 0.f, 0.f, 0.f, 0.f};
    const int ra = full ? (row0 + l) : min(row0 + l, N - 1);

    #pragma unroll 4
    for (int k0 = 0; k0 < IN_FT; k0 += 4) {
        const int ka = k0 + 2 * hi;
        v2f a, b;
        a.x = h[(size_t)ra * IN_FT + ka];
        a.y = h[(size_t)ra * IN_FT + ka + 1];
        b.x = W[(size_t)ka * HD + col0 + l];
        b.y = W[(size_t)(ka + 1) * HD + col0 + l];
        c = __builtin_amdgcn_wmma_f32_16x16x4_f32(
                false, a, false, b, (short)0, c, false, false);
    }
    if (full) {
        #pragma unroll
        for (int r = 0; r < 8; ++r)
            feat[(size_t)(row0 + r + 8 * hi) * HD + col0 + l] = c[r];
    } else {
        #pragma unroll
        for (int r = 0; r < 8; ++r) {
            const int row = row0 + r + 8 * hi;
            if (row < N) feat[(size_t)row * HD + col0 + l] = c[r];
        }
    }
}

// ---------------------------------------------------------------------------
// el[n,h] = feat[n,h,:] . attn_l[h,:]   er likewise   (one thread per (n,h))
// ---------------------------------------------------------------------------
__global__ void node_logits_kernel(
    const float* __restrict__ feat,
    const float* __restrict__ attn_l, const float* __restrict__ attn_r,
    float* __restrict__ el, float* __restrict__ er, int total)
{
    const int i = blockIdx.x * blockDim.x + threadIdx.x;
    if (i >= total) return;
    const int n = i >> 2, hh = i & 3;
    const float* f  = feat + (size_t)n * HD + hh * DHEAD;
    const float* al = attn_l + hh * DHEAD;
    const float* ar = attn_r + hh * DHEAD;
    float sl = 0.f, sr = 0.f;
    #pragma unroll
    for (int d = 0; d < DHEAD; ++d) { sl += f[d] * al[d]; sr += f[d] * ar[d]; }
    el[i] = sl;
    er[i] = sr;
}

// ---------------------------------------------------------------------------
// We_fold[k, n] = (n<4) ? sum_d We[k, n*16+d] * attn_e[n,d] : 0   [64 x 16]
// ---------------------------------------------------------------------------
__global__ void fold_we_kernel(
    const float* __restrict__ We, const float* __restrict__ attn_e,
    float* __restrict__ wfold)
{
    const int i = blockIdx.x * blockDim.x + threadIdx.x;
    if (i >= 64 * 16) return;
    const int k = i >> 4, n = i & 15;
    float s = 0.f;
    if (n < NHEAD) {
        #pragma unroll
        for (int d = 0; d < DHEAD; ++d)
            s += We[(size_t)k * HD + n * DHEAD + d] * attn_e[n * DHEAD + d];
    }
    wfold[i] = s;
}

// ---------------------------------------------------------------------------
// ee = edge_feat[E,64] @ We_fold[64,16] via WMMA (cols 0..3 valid), fused
// with logit = leakyrelu(el[src] + er[dst] + ee) and float atomic segment-max.
// One wave per 16-edge tile; 8 waves per block.
// ---------------------------------------------------------------------------
__global__ __launch_bounds__(256) void ee_logit_kernel(
    const float* __restrict__ ef, const float* __restrict__ wfold,
    const int* __restrict__ src, const int* __restrict__ dst,
    const float* __restrict__ el, const float* __restrict__ er,
    float* __restrict__ ebuf, float* __restrict__ mmax, int E)
{
    const int wave = threadIdx.x >> 5;
    const int lane = threadIdx.x & 31;
    const int hi   = lane >> 4;
    const int l    = lane & 15;
    const int tile = blockIdx.x * 8 + wave;
    const int e0   = tile * 16;
    if (e0 >= E) return;
    const bool full = (e0 + 16 <= E);

    v8f c = {0.f, 0.f, 0.f, 0.f, 0.f, 0.f, 0.f, 0.f};
    const int ra = full ? (e0 + l) : min(e0 + l, E - 1);

    #pragma unroll 4
    for (int k0 = 0; k0 < HD; k0 += 4) {
        const int ka = k0 + 2 * hi;
        v2f a, b;
        a.x = ef[(size_t)ra * HD + ka];
        a.y = ef[(size_t)ra * HD + ka + 1];
        b.x = wfold[ka * 16 + l];
        b.y = wfold[(ka + 1) * 16 + l];
        c = __builtin_amdgcn_wmma_f32_16x16x4_f32(
                false, a, false, b, (short)0, c, false, false);
    }
    if (l < NHEAD) {                        // column l = head index
        #pragma unroll
        for (int r = 0; r < 8; ++r) {
            const int e = e0 + r + 8 * hi;
            if (!full && e >= E) continue;
            const int s = src[e], d2 = dst[e];
            float v = el[s * NHEAD + l] + er[d2 * NHEAD + l] + c[r];
            v = (v > 0.f) ? v : SLOPE * v;  // leaky relu
            ebuf[(size_t)e * NHEAD + l] = v;
            atomic_max_float(&mmax[d2 * NHEAD + l], v);
        }
    }
}

// ---------------------------------------------------------------------------
// ex = exp(e - m[dst]); denom[dst,h] += ex   (thread per (e,h))
// ---------------------------------------------------------------------------
__global__ void exp_denom_kernel(
    float* __restrict__ ebuf, const int* __restrict__ dst,
    const float* __restrict__ mmax, float* __restrict__ denom, int total)
{
    const int i = blockIdx.x * blockDim.x + threadIdx.x;
    if (i >= total) return;
    const int e = i >> 2, hh = i & 3;
    const int d2 = dst[e];
    const float ex = expf(ebuf[i] - mmax[d2 * NHEAD + hh]);
    ebuf[i] = ex;
    atomicAdd(&denom[d2 * NHEAD + hh], ex);
}

// ---------------------------------------------------------------------------
// out[dst,h,d] += (ex / max(denom[dst,h],1e-16)) * feat[src,h,d]
// 64 threads per edge (t = h*16+d); out & feat are L2-resident.
// ---------------------------------------------------------------------------
__global__ __launch_bounds__(256) void aggregate_kernel(
    const float* __restrict__ ebuf, const float* __restrict__ denom,
    const float* __restrict__ feat,
    const int* __restrict__ src, const int* __restrict__ dst,
    float* __restrict__ out, int E)
{
    const int e = blockIdx.x * 4 + (threadIdx.x >> 6);
    if (e >= E) return;
    const int t  = threadIdx.x & 63;
    const int hh = t >> 4;
    const int s = src[e], d2 = dst[e];
    const float ex = ebuf[(size_t)e * NHEAD + hh];
    const float dn = denom[d2 * NHEAD + hh];
    const float alpha = ex / fmaxf(dn, 1e-16f);
    atomicAdd(&out[(size_t)d2 * HD + t], alpha * feat[(size_t)s * HD + t]);
}

// ---------------------------------------------------------------------------
// ELU in place
// ---------------------------------------------------------------------------
__global__ void elu_kernel(float* __restrict__ out, int total)
{
    const int i = blockIdx.x * blockDim.x + threadIdx.x;
    if (i >= total) return;
    const float x = out[i];
    out[i] = (x > 0.f) ? x : expm1f(x);
}

// ---------------------------------------------------------------------------
extern "C" void kernel_launch(void* const* d_in, const int* in_sizes, int n_in,
                              void* d_out, int out_size, void* d_ws, size_t ws_size,
                              hipStream_t stream)
{
    (void)n_in; (void)ws_size;
    const float* h         = (const float*)d_in[0];
    const float* edge_feat = (const float*)d_in[1];
    const int*   src       = (const int*)d_in[2];
    const int*   dst       = (const int*)d_in[3];
    const float* W         = (const float*)d_in[4];
    const float* We        = (const float*)d_in[5];
    const float* attn_l    = (const float*)d_in[6];
    const float* attn_r    = (const float*)d_in[7];
    const float* attn_e    = (const float*)d_in[8];
    float* out = (float*)d_out;

    const int N = in_sizes[0] / IN_FT;
    const int E = in_sizes[2];

    // workspace carve-out (256B aligned)
    char* ws = (char*)d_ws;
    size_t off = 0;
    auto carve = [&](size_t bytes) -> void* {
        void* p = ws + off;
        off += (bytes + 255) & ~(size_t)255;
        return p;
    };
    float* feat  = (float*)carve((size_t)N * HD * sizeof(float));
    float* el    = (float*)carve((size_t)N * NHEAD * sizeof(float));
    float* er    = (float*)carve((size_t)N * NHEAD * sizeof(float));
    float* wfold = (float*)carve(64 * 16 * sizeof(float));
    float* ebuf  = (float*)carve((size_t)E * NHEAD * sizeof(float));
    float* mmax  = (float*)carve((size_t)N * NHEAD * sizeof(float));
    float* denom = (float*)carve((size_t)N * NHEAD * sizeof(float));

    hipMemsetAsync(mmax,  0xFF, (size_t)N * NHEAD * sizeof(float), stream); // -NaN bits: both atomic paths overwrite
    hipMemsetAsync(denom, 0x00, (size_t)N * NHEAD * sizeof(float), stream);
    hipMemsetAsync(out,   0x00, (size_t)out_size * sizeof(float), stream);

    const int rowTiles = (N + 15) / 16;
    feat_gemm_kernel<<<rowTiles, 128, 0, stream>>>(h, W, feat, N);

    node_logits_kernel<<<(N * NHEAD + 255) / 256, 256, 0, stream>>>(
        feat, attn_l, attn_r, el, er, N * NHEAD);

    fold_we_kernel<<<4, 256, 0, stream>>>(We, attn_e, wfold);

    const int eTiles = (E + 15) / 16;
    ee_logit_kernel<<<(eTiles + 7) / 8, 256, 0, stream>>>(
        edge_feat, wfold, src, dst, el, er, ebuf, mmax, E);

    exp_denom_kernel<<<(E * NHEAD + 255) / 256, 256, 0, stream>>>(
        ebuf, dst, mmax, denom, E * NHEAD);

    aggregate_kernel<<<(E + 3) / 4, 256, 0, stream>>>(
        ebuf, denom, feat, src, dst, out, E);

    elu_kernel<<<(N * HD + 255) / 256, 256, 0, stream>>>(out, N * HD);
}